// Conv1x1_53429393162852
// MI455X (gfx1250) — compile-verified
//
#include <hip/hip_runtime.h>
#include <cstdint>

// Problem: x (8,64,56,56) f32, w (64,64) f32 -> out (8,56,56 per co) f32
// out[b,co,h,w] = max over ci of |x[b,ci,h,w] + w[co,ci]|
//
// CIN = COUT = 64, HW = 56*56 = 3136, per-batch x plane = 64*3136 = 200704.

#define HW        3136
#define PLANE     200704     // 64 * 3136
#define PIX_BLK   32         // pixels per block
#define CO_THR    8          // output channels per thread
#define NTHREADS  256        // 32 pixels * 8 co-groups

__global__ __launch_bounds__(NTHREADS)
void maxabs_pw_kernel(const float* __restrict__ x,
                      const float* __restrict__ wgt,
                      float* __restrict__ out)
{
    // LDS tile: smem[ci][pix]  (64 x 32 floats = 8 KB), [ci]-major so that
    // per-lane reads (lane == pix) are bank-conflict-free.
    __shared__ float smem[64 * PIX_BLK];

    const int tid = threadIdx.x;
    const int P0  = blockIdx.x * PIX_BLK;   // base flat pixel (b*HW + hw)
    const int b   = P0 / HW;                // 3136 % 32 == 0 -> no b crossing
    const int hw0 = P0 - b * HW;

    // Base of this block's pixel window inside batch b (add ci*HW + p per elem)
    const float* xbase = x + (size_t)b * PLANE + hw0;

    // ---- Stage 32 pixels x 64 channels (2048 dwords) via async copy to LDS.
    // 8 elements per thread; each 32-lane run covers one ci with p = lane,
    // i.e. one fully coalesced 128B global line per instruction.
    #pragma unroll
    for (int k = 0; k < 8; ++k) {
        const int e  = tid + k * NTHREADS;
        const int ci = e >> 5;
        const int p  = e & 31;
        const uint32_t goff = (uint32_t)((ci * HW + p) * 4);
        const uint32_t loff =
            (uint32_t)(uintptr_t)(&smem[ci * PIX_BLK + p]); // low 32b = LDS addr
        asm volatile("global_load_async_to_lds_b32 %0, %1, %2"
                     :
                     : "v"(loff), "v"(goff), "s"(xbase)
                     : "memory");
    }
    asm volatile("s_wait_asynccnt 0" ::: "memory");
    __syncthreads();

    const int pix = tid & 31;               // == lane id within the wave
    // tid>>5 is wave-uniform; readfirstlane tells the compiler, so the weight
    // row loads below become scalar s_load and feed the SGPR operand of
    // v_add_f32 (2 VALU ops per (ci,co), no per-lane weight traffic).
    const int cog = __builtin_amdgcn_readfirstlane(tid >> 5);

    // Pull this pixel's 64 channel values into registers (one LDS read each).
    float xv[64];
    #pragma unroll
    for (int ci = 0; ci < 64; ++ci)
        xv[ci] = smem[ci * PIX_BLK + pix];

    const int obase = b * PLANE + hw0 + pix;

    for (int j = 0; j < CO_THR; ++j) {
        const int co = cog * CO_THR + j;
        const float* __restrict__ wrow = wgt + co * 64;  // uniform -> s_load
        float acc = 0.0f;                                // |t| >= 0, so 0 is safe
        #pragma unroll
        for (int ci = 0; ci < 64; ++ci) {
            const float t = xv[ci] + wrow[ci];           // v_add_f32 v, s, v
            acc = fmaxf(acc, fabsf(t));                  // v_max_f32 acc, acc, |t|
        }
        out[obase + co * HW] = acc;                      // coalesced store
    }
}

extern "C" void kernel_launch(void* const* d_in, const int* in_sizes, int n_in,
                              void* d_out, int out_size, void* d_ws, size_t ws_size,
                              hipStream_t stream)
{
    (void)in_sizes; (void)n_in; (void)d_ws; (void)ws_size; (void)out_size;
    const float* x   = (const float*)d_in[0];   // (8,64,56,56) f32
    const float* wgt = (const float*)d_in[1];   // (64,64,1,1)  f32
    float* out       = (float*)d_out;           // (8,64,56,56) f32

    const int total_pixels = 8 * HW;            // 25088
    const int blocks = total_pixels / PIX_BLK;  // 784, exact
    maxabs_pw_kernel<<<blocks, NTHREADS, 0, stream>>>(x, wgt, out);
}